// RNNDecoder_5446018531729
// MI455X (gfx1250) — compile-verified
//
#include <hip/hip_runtime.h>
#include <math.h>

#define VOCAB   50000
#define EMB     128
#define HID     128
#define BEAM    4
#define MAXLEN  24
#define BATCH   16
#define TENC    64
#define START_IDX 2
#define NROW    (BEAM*BATCH)   /* 64 */
#define FEATD   (2*HID)        /* 256 */
#define NTILES  (VOCAB/16)     /* 3125 */

typedef float v2f __attribute__((ext_vector_type(2)));
typedef float v8f __attribute__((ext_vector_type(8)));

#if defined(__has_builtin)
#if __has_builtin(__builtin_amdgcn_wmma_f32_16x16x4_f32)
#define HAS_WMMA_F32 1
#endif
#endif

// ---------------------------------------------------------------------------
// Init: h_cur[k,b,:] = initial_hidden[0,b,:]; last_pred=START; lps=0
// ---------------------------------------------------------------------------
__global__ void __launch_bounds__(256) init_state_kernel(
    const float* __restrict__ init_h, float* __restrict__ h_cur,
    float* __restrict__ lps, int* __restrict__ last_pred)
{
  int i = blockIdx.x * 256 + threadIdx.x;
  if (i < NROW * HID) {
    int b = (i / HID) % BATCH;
    int c = i % HID;
    h_cur[i] = init_h[b * HID + c];
  }
  if (i < NROW) { lps[i] = 0.0f; last_pred[i] = START_IDX; }
}

// ---------------------------------------------------------------------------
// GRU cell + dot attention, one block per (k,b) row. 128 threads.
// ---------------------------------------------------------------------------
__global__ void __launch_bounds__(128) gru_attn_kernel(
    const float* __restrict__ enc,      // [16,64,256]
    const float* __restrict__ emb_tab,  // [VOCAB,128]
    const float* __restrict__ W_ih,     // [384,128]
    const float* __restrict__ W_hh,     // [384,128]
    const float* __restrict__ b_ih, const float* __restrict__ b_hh,
    const int*   __restrict__ last_pred,
    const float* __restrict__ h_cur,
    float* __restrict__ h_new,
    float* __restrict__ feat_step)      // [64,256] for this step
{
  __shared__ float s_x[EMB], s_h[HID], s_hn[HID], s_p[TENC];
  const int tid = threadIdx.x;
  const int row = blockIdx.x;          // k*16 + b
  const int b   = row & 15;

  int tok = last_pred[row];
  s_x[tid] = emb_tab[(size_t)tok * EMB + tid];
  s_h[tid] = h_cur[row * HID + tid];
  __syncthreads();

  float gir = b_ih[tid], giz = b_ih[HID + tid], gin = b_ih[2 * HID + tid];
  float ghr = b_hh[tid], ghz = b_hh[HID + tid], ghn = b_hh[2 * HID + tid];
  const float* wi0 = W_ih + (size_t)tid * EMB;
  const float* wi1 = W_ih + (size_t)(HID + tid) * EMB;
  const float* wi2 = W_ih + (size_t)(2 * HID + tid) * EMB;
  const float* wh0 = W_hh + (size_t)tid * HID;
  const float* wh1 = W_hh + (size_t)(HID + tid) * HID;
  const float* wh2 = W_hh + (size_t)(2 * HID + tid) * HID;
  for (int e = 0; e < EMB; ++e) {
    float xe = s_x[e], he = s_h[e];
    gir += xe * wi0[e];  giz += xe * wi1[e];  gin += xe * wi2[e];
    ghr += he * wh0[e];  ghz += he * wh1[e];  ghn += he * wh2[e];
  }
  float r = 1.0f / (1.0f + expf(-(gir + ghr)));
  float z = 1.0f / (1.0f + expf(-(giz + ghz)));
  float n = tanhf(gin + r * ghn);
  float hn = (1.0f - z) * n + z * s_h[tid];
  s_hn[tid] = hn;
  h_new[row * HID + tid] = hn;
  __syncthreads();

  // attention scores over TENC forward encoder states
  if (tid < TENC) {
    const float* er = enc + (size_t)b * TENC * FEATD + (size_t)tid * FEATD;
    float sc = 0.0f;
    for (int h = 0; h < HID; ++h) sc += er[h] * s_hn[h];
    s_p[tid] = sc;
  }
  __syncthreads();
  if (tid == 0) {
    float m = s_p[0];
    for (int t = 1; t < TENC; ++t) m = fmaxf(m, s_p[t]);
    float s = 0.0f;
    for (int t = 0; t < TENC; ++t) { float e = expf(s_p[t] - m); s_p[t] = e; s += e; }
    float inv = 1.0f / s;
    for (int t = 0; t < TENC; ++t) s_p[t] *= inv;
  }
  __syncthreads();
  float at = 0.0f;
  for (int t = 0; t < TENC; ++t)
    at += s_p[t] * enc[(size_t)b * TENC * FEATD + (size_t)t * FEATD + tid];

  float* fr = feat_step + (size_t)row * FEATD;
  fr[tid]       = at;    // attn part
  fr[HID + tid] = hn;    // hidden part
}

// ---------------------------------------------------------------------------
// WMMA GEMM: out[row, n] = sum_c A[row,c]*W_logit[n,c] + b_logit[n] (- bias[row])
// M multiple of 64. One wave per 16-column tile; 8 waves/block.
// Output stride is the compile-time constant VOCAB so row offsets fold into
// the 24-bit VMEM instruction offset; bias handling is a template parameter
// so the hot per-step path has a branch-free epilogue.
// ---------------------------------------------------------------------------
template <bool HAS_BIAS>
__global__ void __launch_bounds__(256) gemm_logits_kernel(
    const float* __restrict__ A, const float* __restrict__ Wl,
    const float* __restrict__ b_logit, const float* __restrict__ bias,
    float* __restrict__ out, int M)
{
  const int lane  = threadIdx.x & 31;
  const int wv    = threadIdx.x >> 5;
  const int ntile = blockIdx.x * 8 + wv;
  if (ntile >= NTILES) return;          // wave-uniform: EXEC stays all-ones
  const int n0   = ntile * 16;
  const int nloc = lane & 15;
  const int half = lane >> 4;
  const float* Bcol = Wl + (size_t)(n0 + nloc) * FEATD + 2 * half;
  const float bl = b_logit[n0 + nloc];
  const int nmt = M >> 4;

  for (int mt0 = 0; mt0 < nmt; mt0 += 4) {
    v8f acc[4];
    #pragma unroll
    for (int m = 0; m < 4; ++m)
      #pragma unroll
      for (int r = 0; r < 8; ++r) acc[m][r] = 0.0f;

#ifdef HAS_WMMA_F32
    for (int kk = 0; kk < FEATD; kk += 4) {
      v2f bfrag = *(const v2f*)(Bcol + kk);
      #pragma unroll
      for (int m = 0; m < 4; ++m) {
        v2f afrag = *(const v2f*)(A + (size_t)((mt0 + m) * 16 + nloc) * FEATD
                                    + kk + 2 * half);
        acc[m] = __builtin_amdgcn_wmma_f32_16x16x4_f32(
            false, afrag, false, bfrag, (short)0, acc[m], false, false);
      }
    }
#else
    // scalar fallback with identical C layout (M = 8*half + r, N = nloc)
    for (int m = 0; m < 4; ++m)
      for (int r = 0; r < 8; ++r) {
        const float* Ar = A + (size_t)((mt0 + m) * 16 + half * 8 + r) * FEATD;
        const float* Wr = Wl + (size_t)(n0 + nloc) * FEATD;
        float s = 0.0f;
        for (int c = 0; c < FEATD; ++c) s += Ar[c] * Wr[c];
        acc[m][r] = s;
      }
#endif
    #pragma unroll
    for (int m = 0; m < 4; ++m) {
      const int row0 = (mt0 + m) * 16 + half * 8;   // 8 consecutive rows
      float* po = out + (size_t)row0 * VOCAB + n0 + nloc;
      if (HAS_BIAS) {
        float bs[8];
        #pragma unroll
        for (int r = 0; r < 8; ++r) bs[r] = bias[row0 + r];
        #pragma unroll
        for (int r = 0; r < 8; ++r)
          po[(size_t)r * VOCAB] = acc[m][r] + bl - bs[r];
      } else {
        #pragma unroll
        for (int r = 0; r < 8; ++r)
          po[(size_t)r * VOCAB] = acc[m][r] + bl;
      }
    }
  }
}

// ---------------------------------------------------------------------------
// Per-row streaming logsumexp + stable top-4 (by logit, tie -> lower index)
// ---------------------------------------------------------------------------
__device__ inline void top4_insert(float* tv, int* ti, float v, int idx)
{
  // strict > : on tie the existing (lower index, scanned earlier) wins
  #pragma unroll
  for (int j = 0; j < 4; ++j) {
    if (v > tv[j]) {
      for (int q = 3; q > j; --q) { tv[q] = tv[q - 1]; ti[q] = ti[q - 1]; }
      tv[j] = v; ti[j] = idx;
      return;
    }
  }
}

__device__ inline void top4_merge(float* av, int* ai, const float* bv, const int* bi)
{
  float ov[4]; int oi[4]; int pa = 0, pb = 0;
  #pragma unroll
  for (int j = 0; j < 4; ++j) {
    bool takeA;
    if (pa >= 4)      takeA = false;
    else if (pb >= 4) takeA = true;
    else takeA = (av[pa] > bv[pb]) || (av[pa] == bv[pb] && ai[pa] <= bi[pb]);
    if (takeA) { ov[j] = av[pa]; oi[j] = ai[pa]; ++pa; }
    else       { ov[j] = bv[pb]; oi[j] = bi[pb]; ++pb; }
  }
  #pragma unroll
  for (int j = 0; j < 4; ++j) { av[j] = ov[j]; ai[j] = oi[j]; }
}

__global__ void __launch_bounds__(256) softmax_topk_kernel(
    const float* __restrict__ logits,   // [64, VOCAB]
    float* __restrict__ rowmax_out, float* __restrict__ logZ_out,
    float* __restrict__ cand_val, int* __restrict__ cand_idx)
{
  __shared__ float s_m[256], s_s[256];
  __shared__ float s_tv[256 * 4];
  __shared__ int   s_ti[256 * 4];
  const int tid = threadIdx.x;
  const int row = blockIdx.x;
  const float* base = logits + (size_t)row * VOCAB;

  float m = -INFINITY, s = 0.0f;
  float tv[4] = {-INFINITY, -INFINITY, -INFINITY, -INFINITY};
  int   ti[4] = {0x7fffffff, 0x7fffffff, 0x7fffffff, 0x7fffffff};
  for (int i = tid; i < VOCAB; i += 256) {
    float x = base[i];
    if (x > m) { s = s * expf(m - x) + 1.0f; m = x; }
    else       { s += expf(x - m); }
    top4_insert(tv, ti, x, i);
  }
  s_m[tid] = m; s_s[tid] = s;
  #pragma unroll
  for (int j = 0; j < 4; ++j) { s_tv[tid * 4 + j] = tv[j]; s_ti[tid * 4 + j] = ti[j]; }

  for (int off = 128; off > 0; off >>= 1) {
    __syncthreads();
    if (tid < off) {
      float m2 = s_m[tid + off], s2 = s_s[tid + off];
      float M = fmaxf(s_m[tid], m2);
      s_s[tid] = s_s[tid] * expf(s_m[tid] - M) + s2 * expf(m2 - M);
      s_m[tid] = M;
      float av[4]; int ai[4];
      #pragma unroll
      for (int j = 0; j < 4; ++j) { av[j] = s_tv[tid * 4 + j]; ai[j] = s_ti[tid * 4 + j]; }
      top4_merge(av, ai, &s_tv[(tid + off) * 4], &s_ti[(tid + off) * 4]);
      #pragma unroll
      for (int j = 0; j < 4; ++j) { s_tv[tid * 4 + j] = av[j]; s_ti[tid * 4 + j] = ai[j]; }
    }
  }
  __syncthreads();
  if (tid == 0) {
    rowmax_out[row] = s_m[0];
    logZ_out[row]   = logf(s_s[0]);
    #pragma unroll
    for (int j = 0; j < 4; ++j) {
      cand_val[row * 4 + j] = s_tv[j];
      cand_idx[row * 4 + j] = s_ti[j];
    }
  }
}

// ---------------------------------------------------------------------------
// Merge 16 candidates per batch -> global top-4, update beams
// ---------------------------------------------------------------------------
__global__ void __launch_bounds__(256) select_kernel(
    const float* __restrict__ cand_val, const int* __restrict__ cand_idx,
    const float* __restrict__ rowmax, const float* __restrict__ logZ,
    float* __restrict__ lps, int* __restrict__ last_pred,
    const float* __restrict__ h_new, float* __restrict__ h_cur,
    int* __restrict__ sel_k_step, int* __restrict__ tok_step)
{
  __shared__ int s_selk[BATCH * BEAM];
  const int tid = threadIdx.x;
  if (tid < BATCH) {
    const int b = tid;
    float bv[4] = {-INFINITY, -INFINITY, -INFINITY, -INFINITY};
    int   bf[4] = {0x7fffffff, 0x7fffffff, 0x7fffffff, 0x7fffffff};
    float oldlps[BEAM];
    #pragma unroll
    for (int k = 0; k < BEAM; ++k) oldlps[k] = lps[k * BATCH + b];
    for (int k = 0; k < BEAM; ++k) {
      int row = k * BATCH + b;
      float sub = rowmax[row] + logZ[row];
      #pragma unroll
      for (int j = 0; j < 4; ++j) {
        float v = cand_val[row * 4 + j] - sub + oldlps[k];
        int fl = k * VOCAB + cand_idx[row * 4 + j];
        // stable insert by (value desc, flat index asc)
        for (int q = 0; q < 4; ++q) {
          if (v > bv[q] || (v == bv[q] && fl < bf[q])) {
            for (int p = 3; p > q; --p) { bv[p] = bv[p - 1]; bf[p] = bf[p - 1]; }
            bv[q] = v; bf[q] = fl;
            break;
          }
        }
      }
    }
    #pragma unroll
    for (int j = 0; j < 4; ++j) {
      int k  = bf[j] / VOCAB;
      int tk = bf[j] % VOCAB;
      sel_k_step[b * 4 + j] = k;
      tok_step[b * 4 + j]   = tk;
      s_selk[b * 4 + j]     = k;
      lps[j * BATCH + b]       = bv[j];
      last_pred[j * BATCH + b] = tk;
    }
  }
  __syncthreads();
  for (int i = tid; i < BEAM * BATCH * HID; i += 256) {
    int j = i / (BATCH * HID);
    int rem = i % (BATCH * HID);
    int b = rem / HID, c = rem % HID;
    h_cur[(j * BATCH + b) * HID + c] =
        h_new[(s_selk[b * 4 + j] * BATCH + b) * HID + c];
  }
}

// ---------------------------------------------------------------------------
// Backtrack: build selected chain, emit token ids, gather feat rows + biases
// ---------------------------------------------------------------------------
__global__ void __launch_bounds__(256) backtrack_kernel(
    const int* __restrict__ sel_k_all,   // [24,16,4]
    const int* __restrict__ tok_all,     // [24,16,4]
    const float* __restrict__ rowmax_all,// [24,64]
    const float* __restrict__ logZ_all,  // [24,64]
    const float* __restrict__ feat_all,  // [24,64,256]
    int* __restrict__ ksel,              // [24,16]
    float* __restrict__ bias_sel,        // [384]
    float* __restrict__ featsel,         // [384,256]
    float* __restrict__ out_prs)         // [24,16]
{
  const int tid = threadIdx.x;
  if (tid < BATCH) {
    const int b = tid;
    int parent = 0;
    for (int t = MAXLEN - 1; t >= 0; --t) {
      int kstar = sel_k_all[t * NROW + b * 4 + parent];
      int tk    = tok_all  [t * NROW + b * 4 + parent];
      ksel[t * BATCH + b] = kstar;
      out_prs[t * BATCH + b] = (float)tk;
      int row = kstar * BATCH + b;
      bias_sel[t * BATCH + b] = rowmax_all[t * NROW + row] + logZ_all[t * NROW + row];
      parent = kstar;
    }
  }
  __syncthreads();
  for (int i = tid; i < MAXLEN * BATCH * FEATD; i += 256) {
    int r = i >> 8;            // t*16 + b
    int c = i & 255;
    int t = r >> 4, b = r & 15;
    int kk = ksel[t * BATCH + b];
    featsel[i] = feat_all[((size_t)t * NROW + kk * BATCH + b) * FEATD + c];
  }
}

// ---------------------------------------------------------------------------
// Host-side orchestration
// ---------------------------------------------------------------------------
extern "C" void kernel_launch(void* const* d_in, const int* in_sizes, int n_in,
                              void* d_out, int out_size, void* d_ws, size_t ws_size,
                              hipStream_t stream)
{
  (void)in_sizes; (void)n_in; (void)out_size; (void)ws_size;
  const float* enc      = (const float*)d_in[0];
  const float* init_h   = (const float*)d_in[1];
  const float* emb      = (const float*)d_in[2];
  const float* W_ih     = (const float*)d_in[3];
  const float* W_hh     = (const float*)d_in[4];
  const float* b_ih     = (const float*)d_in[5];
  const float* b_hh     = (const float*)d_in[6];
  const float* W_logit  = (const float*)d_in[7];
  const float* b_logit  = (const float*)d_in[8];
  float* out = (float*)d_out;

  char* ws = (char*)d_ws;
  size_t off = 0;
  auto carve = [&](size_t bytes) -> void* {
    void* p = ws + off;
    off = (off + bytes + 255) & ~(size_t)255;
    return p;
  };

  float* feat_all   = (float*)carve(sizeof(float) * MAXLEN * NROW * FEATD);
  float* h_cur      = (float*)carve(sizeof(float) * NROW * HID);
  float* h_new      = (float*)carve(sizeof(float) * NROW * HID);
  float* lps        = (float*)carve(sizeof(float) * NROW);
  int*   last_pred  = (int*)  carve(sizeof(int)   * NROW);
  float* logits_buf = (float*)carve(sizeof(float) * (size_t)NROW * VOCAB);
  float* rowmax_all = (float*)carve(sizeof(float) * MAXLEN * NROW);
  float* logZ_all   = (float*)carve(sizeof(float) * MAXLEN * NROW);
  float* cand_val   = (float*)carve(sizeof(float) * NROW * 4);
  int*   cand_idx   = (int*)  carve(sizeof(int)   * NROW * 4);
  int*   sel_k_all  = (int*)  carve(sizeof(int)   * MAXLEN * NROW);
  int*   tok_all    = (int*)  carve(sizeof(int)   * MAXLEN * NROW);
  int*   ksel       = (int*)  carve(sizeof(int)   * MAXLEN * BATCH);
  float* bias_sel   = (float*)carve(sizeof(float) * MAXLEN * BATCH);
  float* featsel    = (float*)carve(sizeof(float) * MAXLEN * BATCH * FEATD);

  const int gemm_blocks = (NTILES + 7) / 8;  // 391

  init_state_kernel<<<(NROW * HID + 255) / 256, 256, 0, stream>>>(
      init_h, h_cur, lps, last_pred);

  for (int t = 0; t < MAXLEN; ++t) {
    gru_attn_kernel<<<NROW, 128, 0, stream>>>(
        enc, emb, W_ih, W_hh, b_ih, b_hh, last_pred, h_cur, h_new,
        feat_all + (size_t)t * NROW * FEATD);

    gemm_logits_kernel<false><<<gemm_blocks, 256, 0, stream>>>(
        feat_all + (size_t)t * NROW * FEATD, W_logit, b_logit,
        /*bias=*/nullptr, logits_buf, NROW);

    softmax_topk_kernel<<<NROW, 256, 0, stream>>>(
        logits_buf, rowmax_all + t * NROW, logZ_all + t * NROW,
        cand_val, cand_idx);

    select_kernel<<<1, 256, 0, stream>>>(
        cand_val, cand_idx, rowmax_all + t * NROW, logZ_all + t * NROW,
        lps, last_pred, h_new, h_cur,
        sel_k_all + t * NROW, tok_all + t * NROW);
  }

  backtrack_kernel<<<1, 256, 0, stream>>>(
      sel_k_all, tok_all, rowmax_all, logZ_all, feat_all,
      ksel, bias_sel, featsel,
      out + (size_t)MAXLEN * BATCH * VOCAB);

  gemm_logits_kernel<true><<<gemm_blocks, 256, 0, stream>>>(
      featsel, W_logit, b_logit, bias_sel, out, MAXLEN * BATCH);
}